// GIN_24953759989866
// MI455X (gfx1250) — compile-verified
//
#include <hip/hip_runtime.h>
#include <hip/hip_bf16.h>
#include <stdint.h>

#define FDIM 128

typedef int v2i32 __attribute__((vector_size(2 * sizeof(int))));

// ---------------------------------------------------------------------------
// CDNA5 async global->LDS copy (ASYNCcnt path) with builtin/asm fallbacks
// ---------------------------------------------------------------------------
__device__ __forceinline__ void async_cp_b64(const void* gptr, void* lptr) {
#if __has_builtin(__builtin_amdgcn_global_load_async_to_lds_b64)
    __builtin_amdgcn_global_load_async_to_lds_b64(
        (__attribute__((address_space(1))) v2i32*)gptr,   // global (AS1)
        (__attribute__((address_space(3))) v2i32*)lptr,   // LDS (AS3)
        0, 0);
#else
    unsigned l = (unsigned)(uintptr_t)(__attribute__((address_space(3))) void*)lptr;
    unsigned long long g = (unsigned long long)(uintptr_t)gptr;
    asm volatile("global_load_async_to_lds_b64 %0, %1, off"
                 :: "v"(l), "v"(g) : "memory");
#endif
}

__device__ __forceinline__ void wait_async_le2() {
#if __has_builtin(__builtin_amdgcn_s_wait_asynccnt)
    __builtin_amdgcn_s_wait_asynccnt(2);
#else
    asm volatile("s_wait_asynccnt 2" ::: "memory");
#endif
}

__device__ __forceinline__ void wait_async_le0() {
#if __has_builtin(__builtin_amdgcn_s_wait_asynccnt)
    __builtin_amdgcn_s_wait_asynccnt(0);
#else
    asm volatile("s_wait_asynccnt 0" ::: "memory");
#endif
}

// ---------------------------------------------------------------------------
// Scatter-add (first in file so its disasm shows in the snippet).
// One wave32 per edge: 32 lanes x float4 == 128 features, one coalesced
// global_load_b128 per edge.  Edge indices are double-buffered into LDS via
// CDNA5 async copies (ASYNCcnt); accumulation uses no-return f32 atomics
// that ride the 192MB L2 (5MB output working set is fully L2-resident).
// ---------------------------------------------------------------------------
__global__ void __launch_bounds__(256)
gin_scatter_kernel(const float* __restrict__ node,
                   const long long* __restrict__ recv,
                   const long long* __restrict__ send,
                   float* __restrict__ out, int nChunks) {
    // [wave][buf][recv/send][lane] : 8 * 2 * 2 * 32 * 8B = 8 KB
    __shared__ long long stage[8][2][2][32];

    const int lane   = threadIdx.x & 31;
    const int wave   = threadIdx.x >> 5;
    const int wpb    = blockDim.x >> 5;
    const int gwave  = blockIdx.x * wpb + wave;
    const int stride = gridDim.x * wpb;

    int c = gwave;
    if (c < nChunks) {
        int e = c * 32 + lane;
        async_cp_b64(&recv[e], &stage[wave][0][0][lane]);
        async_cp_b64(&send[e], &stage[wave][0][1][lane]);
    }

    int buf = 0;
    for (; c < nChunks; c += stride) {
        int cn = c + stride;
        if (cn < nChunks) {
            int e = cn * 32 + lane;
            async_cp_b64(&recv[e], &stage[wave][buf ^ 1][0][lane]);
            async_cp_b64(&send[e], &stage[wave][buf ^ 1][1][lane]);
            wait_async_le2();   // current buffer's 2 async ops retired
        } else {
            wait_async_le0();
        }

#pragma unroll 8
        for (int k = 0; k < 32; ++k) {
            // wave-uniform row indices -> promote to SGPRs (saddr-form loads)
            int r = __builtin_amdgcn_readfirstlane((int)stage[wave][buf][0][k]);
            int s = __builtin_amdgcn_readfirstlane((int)stage[wave][buf][1][k]);
            const float4 v = *(const float4*)(node + s * FDIM + lane * 4);
            float* o = out + r * FDIM + lane * 4;
            atomicAdd(o + 0, v.x);   // no-return global_atomic_add_f32
            atomicAdd(o + 1, v.y);
            atomicAdd(o + 2, v.z);
            atomicAdd(o + 3, v.w);
        }
        buf ^= 1;
    }
}

// ---------------------------------------------------------------------------
// out = (1 + eps) * node   (float4 vectorized; initializes poisoned output)
// ---------------------------------------------------------------------------
__global__ void __launch_bounds__(256)
gin_scale_kernel(const float* __restrict__ node, const float* __restrict__ eps,
                 float* __restrict__ out, int n4) {
    int i = blockIdx.x * blockDim.x + threadIdx.x;
    if (i >= n4) return;
    float s = 1.0f + eps[0];
    float4 v = ((const float4*)node)[i];
    v.x *= s; v.y *= s; v.z *= s; v.w *= s;
    ((float4*)out)[i] = v;
}

// Tail for edge counts not divisible by 32 (generic safety; 640000 % 32 == 0).
__global__ void __launch_bounds__(128)
gin_tail_kernel(const float* __restrict__ node,
                const long long* __restrict__ recv,
                const long long* __restrict__ send,
                float* __restrict__ out, int start, int nEdges) {
    int f = threadIdx.x;
    for (int e = start + blockIdx.x; e < nEdges; e += gridDim.x) {
        int r = (int)recv[e];
        int s = (int)send[e];
        atomicAdd(&out[r * FDIM + f], node[s * FDIM + f]);
    }
}

// ---------------------------------------------------------------------------
extern "C" void kernel_launch(void* const* d_in, const int* in_sizes, int n_in,
                              void* d_out, int out_size, void* d_ws, size_t ws_size,
                              hipStream_t stream) {
    const float*     node = (const float*)d_in[0];
    const long long* ei   = (const long long*)d_in[1];
    const float*     eps  = (const float*)d_in[2];
    float*           out  = (float*)d_out;

    const int nEdges = in_sizes[1] / 2;          // edge_index is (2, M)
    const long long* recv = ei;                  // edge_index[0] = receivers
    const long long* send = ei + nEdges;         // edge_index[1] = senders

    // Phase 1: out = (1 + eps) * node
    int n4 = in_sizes[0] / 4;
    gin_scale_kernel<<<(n4 + 255) / 256, 256, 0, stream>>>(node, eps, out, n4);

    // Phase 2: scatter-add messages, 32-edge chunks per wave.
    // Size grid so each wave gets ~2 chunks (perfect balance at 640K edges:
    // 20000 chunks / 10000 waves = exactly 2, pipelined by double buffering).
    int nChunks = nEdges / 32;
    if (nChunks > 0) {
        int blocks = (nChunks + 15) / 16;        // 8 waves/block * 2 chunks/wave
        if (blocks > 8192) blocks = 8192;
        gin_scatter_kernel<<<blocks, 256, 0, stream>>>(node, recv, send, out, nChunks);
    }
    int rem = nEdges - nChunks * 32;
    if (rem > 0) {
        gin_tail_kernel<<<rem, 128, 0, stream>>>(node, recv, send, out,
                                                 nChunks * 32, nEdges);
    }
}